// MyModel_87522843559166
// MI455X (gfx1250) — compile-verified
//
#include <hip/hip_runtime.h>

typedef float v2f __attribute__((ext_vector_type(2)));
typedef float v8f __attribute__((ext_vector_type(8)));

#define BN_EPS 1e-5f

// One wave32 computes a 16x16 output tile of  out = X * (W*inv) + shift
// using V_WMMA_F32_16X16X4_F32, K=10 padded to 12 (3 chunks of K=4).
// Bandwidth-bound: 320 MB total traffic -> ~13.7 us floor at 23.3 TB/s.
__global__ __launch_bounds__(256) void
gemm10_bn_wmma(const float* __restrict__ x,
               const float* __restrict__ w,
               const float* __restrict__ bias,
               const float* __restrict__ gamma,
               const float* __restrict__ beta,
               const float* __restrict__ mmean,
               const float* __restrict__ mvar,
               float* __restrict__ out,
               int rows, int ntiles)
{
    const int lane = threadIdx.x & 31;
    const int n    = lane & 15;       // output column owned by this lane (C/D layout)
    const bool lo  = lane < 16;       // lane group (K-halves for A/B fragments)

    // Fold BatchNorm(inference) + bias into the GEMM:
    //   out = X*W*inv + (bias*inv + beta - mean*inv)
    float inv = 0.0f, shiftp = 0.0f;
    if (n < 10) {
        float iv = gamma[n] * rsqrtf(mvar[n] + BN_EPS);
        inv    = iv;
        shiftp = bias[n] * iv + beta[n] - mmean[n] * iv;
    }

    // B fragments for 3 K-chunks (4x16 f32 each, 2 VGPRs/lane):
    //  lanes 0-15: K = c*4 + {0,1};  lanes 16-31: K = c*4 + {2,3};  N = lane%16
    v2f bf[3];
    #pragma unroll
    for (int c = 0; c < 3; ++c) {
        #pragma unroll
        for (int v = 0; v < 2; ++v) {
            const int k = c * 4 + v + (lo ? 0 : 2);
            float val = 0.0f;
            if (k < 10 && n < 10) val = w[k * 10 + n] * inv;   // fold inv into W
            bf[c][v] = val;
        }
    }

    const int wavesPerBlock = blockDim.x >> 5;
    const int waveId = blockIdx.x * wavesPerBlock + (threadIdx.x >> 5);
    const int nWaves = gridDim.x * wavesPerBlock;

    for (int tile = waveId; tile < ntiles; tile += nWaves) {   // wave-uniform loop
        const int row0 = tile * 16;
        const int arow = row0 + n;          // A-fragment row owned by this lane

        if (row0 + 16 <= rows) {
            // ---------- full-tile fast path (steady state) ----------
            // A fragment addresses: one per-lane base, imm offsets for the rest.
            const float* ap  = x + (long)arow * 10 + (lo ? 0 : 2); // K chunk base
            const float* ap2 = x + (long)arow * 10 + (lo ? 8 : 0); // K=8,9 (hi: safe addr)

            v2f a0 = __builtin_nontemporal_load((const v2f*)ap);        // K 0,1 / 2,3
            v2f a1 = __builtin_nontemporal_load((const v2f*)(ap + 4));  // K 4,5 / 6,7
            v2f t2 = __builtin_nontemporal_load((const v2f*)ap2);       // K 8,9 (lo)
            v2f a2;                                                     // hi lanes: K=10,11 pad
            a2[0] = lo ? t2[0] : 0.0f;   // branchless zero (v_cndmask), keeps EXEC intact
            a2[1] = lo ? t2[1] : 0.0f;

            // Accumulator pre-loaded with the folded shift (same N for all 8 regs).
            v8f acc;
            #pragma unroll
            for (int i = 0; i < 8; ++i) acc[i] = shiftp;

            // EXEC is all-1s here (no predication above used exec branches).
            acc = __builtin_amdgcn_wmma_f32_16x16x4_f32(false, a0, false, bf[0],
                                                        (short)0, acc, false, false);
            acc = __builtin_amdgcn_wmma_f32_16x16x4_f32(false, a1, false, bf[1],
                                                        (short)0, acc, false, false);
            acc = __builtin_amdgcn_wmma_f32_16x16x4_f32(false, a2, false, bf[2],
                                                        (short)0, acc, false, false);

            // C/D layout: VGPR r -> (M = r (lanes<16) or r+8, N = lane%16).
            // Single exec mask for the lane-constant n<10 predicate; all 8 stores
            // share one base pointer and use immediate offsets (r*40 bytes).
            if (n < 10) {
                float* op = out + (long)(row0 + (lo ? 0 : 8)) * 10 + n;
                #pragma unroll
                for (int r = 0; r < 8; ++r)
                    __builtin_nontemporal_store(acc[r], op + r * 10);
            }
        } else {
            // ---------- partial tail tile (rare / never for rows%16==0) ----------
            const bool rok  = arow < rows;
            const float* rp = x + (long)arow * 10;
            const int koff  = lo ? 0 : 2;

            v2f a0 = {0.0f, 0.0f}, a1 = {0.0f, 0.0f}, a2 = {0.0f, 0.0f};
            if (rok) {
                a0 = __builtin_nontemporal_load((const v2f*)(rp + koff));
                a1 = __builtin_nontemporal_load((const v2f*)(rp + 4 + koff));
                if (lo)
                    a2 = __builtin_nontemporal_load((const v2f*)(rp + 8));
            }

            v8f acc;
            #pragma unroll
            for (int i = 0; i < 8; ++i) acc[i] = shiftp;

            acc = __builtin_amdgcn_wmma_f32_16x16x4_f32(false, a0, false, bf[0],
                                                        (short)0, acc, false, false);
            acc = __builtin_amdgcn_wmma_f32_16x16x4_f32(false, a1, false, bf[1],
                                                        (short)0, acc, false, false);
            acc = __builtin_amdgcn_wmma_f32_16x16x4_f32(false, a2, false, bf[2],
                                                        (short)0, acc, false, false);

            if (n < 10) {
                #pragma unroll
                for (int r = 0; r < 8; ++r) {
                    const int orow = row0 + (lo ? r : r + 8);
                    if (orow < rows)
                        __builtin_nontemporal_store(acc[r], out + (long)orow * 10 + n);
                }
            }
        }
    }
}

extern "C" void kernel_launch(void* const* d_in, const int* in_sizes, int n_in,
                              void* d_out, int out_size, void* d_ws, size_t ws_size,
                              hipStream_t stream)
{
    const float* x     = (const float*)d_in[0];
    const float* w     = (const float*)d_in[1];
    const float* bias  = (const float*)d_in[2];
    const float* gamma = (const float*)d_in[3];
    const float* beta  = (const float*)d_in[4];
    const float* mmean = (const float*)d_in[5];
    const float* mvar  = (const float*)d_in[6];
    float* out = (float*)d_out;

    const int rows   = in_sizes[0] / 10;          // 4,000,000
    const int ntiles = (rows + 15) / 16;          // 250,000 wave-tiles

    const int wavesPerBlock = 8;                  // 256 threads = 8 wave32s
    long want = ((long)ntiles + wavesPerBlock - 1) / wavesPerBlock;
    int blocks = (int)(want > 32768 ? 32768 : want);
    if (blocks < 1) blocks = 1;

    gemm10_bn_wmma<<<blocks, 256, 0, stream>>>(x, w, bias, gamma, beta, mmean,
                                               mvar, out, rows, ntiles);
}